// LocationSensitiveAttentionLayer_2439541424832
// MI455X (gfx1250) — compile-verified
//
#include <hip/hip_runtime.h>

#define B_ 32
#define T_ENC_ 1024
#define T_DEC_ 512
#define E_ 512
#define UNITS_ 512
#define FILTERS_ 32
#define KERNEL_ 31

typedef __attribute__((ext_vector_type(2))) float v2f;
typedef __attribute__((ext_vector_type(8))) float v8f;

__device__ __forceinline__ float fast_tanhf(float x) {
#if __has_builtin(__builtin_amdgcn_tanhf)
    return __builtin_amdgcn_tanhf(x);
#else
    return tanhf(x);
#endif
}

// ---------------------------------------------------------------------------
// Kernel 0: fold conv weights through the location dense layer (once).
//   wc[k,u]  = sum_f conv_w[k,0,f] * w_loc[f,u]   (k in [0,31), row 31 = 0)
//   cbp[u]   = b_a[u] + sum_f conv_b[f] * w_loc[f,u]
// ---------------------------------------------------------------------------
__global__ void lsa_prep_kernel(const float* __restrict__ conv_w,
                                const float* __restrict__ conv_b,
                                const float* __restrict__ w_loc,
                                const float* __restrict__ b_a,
                                float* __restrict__ wc,
                                float* __restrict__ cbp) {
    const int u = threadIdx.x;            // 0..511
    float cb = b_a[u];
    for (int f = 0; f < FILTERS_; ++f)
        cb += conv_b[f] * w_loc[f * UNITS_ + u];
    cbp[u] = cb;
    for (int k = 0; k < KERNEL_; ++k) {
        float s = 0.f;
        for (int f = 0; f < FILTERS_; ++f)
            s += conv_w[k * FILTERS_ + f] * w_loc[f * UNITS_ + u];
        wc[k * UNITS_ + u] = s;
    }
    wc[KERNEL_ * UNITS_ + u] = 0.f;       // K padded to 32 with a zero row
}

// ---------------------------------------------------------------------------
// Kernel 1: persistent per-batch sequential scan over T_DEC steps.
// One workgroup (1024 threads = 32 waves) per batch element.
// Per step: loc = A(shifted align, 1024x32) @ wc(32x512) via WMMA f32 16x16x4,
// fused epilogue e[t] = sum_u v_a[u]*tanh(loc + enc + q + cbp), softmax,
// emit e, cumulate alignment in LDS. wc lives in LDS (the enc stream would
// thrash it out of the WGP$ every step otherwise).
// ---------------------------------------------------------------------------
#define ALIGN_SH_SZ (T_ENC_ + 32)   // A[t,k] = align[t+k-15] -> align_sh[t+k]

__global__ __launch_bounds__(1024)
void lsa_scan_kernel(const float* __restrict__ enc,      // (B, T_enc, E)
                     const float* __restrict__ dec,      // (B, T_dec, E)
                     const float* __restrict__ v_a,      // (U)
                     const float* __restrict__ wc,       // (32, U)
                     const float* __restrict__ cbp,      // (U)
                     float* __restrict__ e_out) {        // (B, T_dec, T_enc)
    __shared__ float wc_sh[32 * UNITS_];     // 64 KB, resident all 512 steps
    __shared__ float align_sh[ALIGN_SH_SZ];  // padded: align_sh[i] = align[i-15]
    __shared__ float qc_sh[UNITS_];
    __shared__ float va_sh[UNITS_];
    __shared__ float e_sh[T_ENC_];
    __shared__ float red_sh[33];

    const int b    = blockIdx.x;
    const int tid  = threadIdx.x;
    const int w    = tid >> 5;               // wave id 0..31
    const int lane = tid & 31;
    const int lo   = lane & 15;
    const int hi   = lane >> 4;

    // one-time staging: wc, v_a, zeroed alignment (incl. +/-15 padding)
    for (int i = tid; i < 32 * UNITS_; i += 1024) wc_sh[i] = wc[i];
    if (tid < UNITS_) va_sh[tid] = v_a[tid];
    if (tid < ALIGN_SH_SZ) align_sh[tid] = 0.f;
    if (tid + 1024 < ALIGN_SH_SZ) align_sh[tid + 1024] = 0.f;
    __syncthreads();

    const float* enc_b = enc + (size_t)b * T_ENC_ * E_;
    const float* dec_b = dec + (size_t)b * T_DEC_ * E_;
    float* eo_b = e_out + (size_t)b * T_DEC_ * T_ENC_;

    for (int step = 0; step < T_DEC_; ++step) {
        // ---- stage query + folded bias; prefetch next query --------------
        if (tid < UNITS_) {
            qc_sh[tid] = dec_b[(size_t)step * E_ + tid] + cbp[tid];
            if (step + 1 < T_DEC_)
                __builtin_prefetch(dec_b + (size_t)(step + 1) * E_ + tid, 0, 1);
        }
        __syncthreads();

        // ---- loc GEMM + fused tanh reduction -----------------------------
        // wave w owns rows t in [32w, 32w+32): two 16-row M tiles processed
        // with dual accumulators so every B fragment feeds two WMMAs.
        const int t0 = 32 * w;
        const int tA = t0 + lo;

        v2f a0[8], a1[8];
#pragma unroll
        for (int kk = 0; kk < 8; ++kk) {
            const int k = 4 * kk + 2 * hi;
            a0[kk].x = align_sh[tA + k];
            a0[kk].y = align_sh[tA + k + 1];
            a1[kk].x = align_sh[tA + 16 + k];
            a1[kk].y = align_sh[tA + 16 + k + 1];
        }

        float acc0[8] = {0.f, 0.f, 0.f, 0.f, 0.f, 0.f, 0.f, 0.f};
        float acc1[8] = {0.f, 0.f, 0.f, 0.f, 0.f, 0.f, 0.f, 0.f};

        for (int j = 0; j < 32; ++j) {         // N tiles over U=512
            const int un = 16 * j + lo;
            v8f c0 = {};
            v8f c1 = {};
#pragma unroll
            for (int kk = 0; kk < 8; ++kk) {
                const int k = 4 * kk + 2 * hi;
                v2f bb;
                bb.x = wc_sh[k * UNITS_ + un];
                bb.y = wc_sh[(k + 1) * UNITS_ + un];
                c0 = __builtin_amdgcn_wmma_f32_16x16x4_f32(
                         false, a0[kk], false, bb, (short)0, c0, false, false);
                c1 = __builtin_amdgcn_wmma_f32_16x16x4_f32(
                         false, a1[kk], false, bb, (short)0, c1, false, false);
            }
            // fused epilogue: tanh + weighted accumulate over u, both tiles
            const float qv = qc_sh[un];
            const float va = va_sh[un];
            const float* ep0 = enc_b + (size_t)(t0 + hi * 8) * E_ + un;
            const float* ep1 = ep0 + (size_t)16 * E_;
#pragma unroll
            for (int r = 0; r < 8; ++r) {
                acc0[r] += va * fast_tanhf(c0[r] + ep0[(size_t)r * E_] + qv);
                acc1[r] += va * fast_tanhf(c1[r] + ep1[(size_t)r * E_] + qv);
            }
        }

        // reduce over the 16 lanes of each half (columns of the tiles)
#pragma unroll
        for (int r = 0; r < 8; ++r) {
            float s0 = acc0[r];
            s0 += __shfl_xor(s0, 1, 32);
            s0 += __shfl_xor(s0, 2, 32);
            s0 += __shfl_xor(s0, 4, 32);
            s0 += __shfl_xor(s0, 8, 32);
            float s1 = acc1[r];
            s1 += __shfl_xor(s1, 1, 32);
            s1 += __shfl_xor(s1, 2, 32);
            s1 += __shfl_xor(s1, 4, 32);
            s1 += __shfl_xor(s1, 8, 32);
            if (lo == 0) {
                e_sh[t0 + hi * 8 + r]      = s0;
                e_sh[t0 + 16 + hi * 8 + r] = s1;
            }
        }
        __syncthreads();

        // ---- softmax over T_enc = 1024 (one value per thread) ------------
        const float x = e_sh[tid];
        float mx = x;
        mx = fmaxf(mx, __shfl_xor(mx, 1, 32));
        mx = fmaxf(mx, __shfl_xor(mx, 2, 32));
        mx = fmaxf(mx, __shfl_xor(mx, 4, 32));
        mx = fmaxf(mx, __shfl_xor(mx, 8, 32));
        mx = fmaxf(mx, __shfl_xor(mx, 16, 32));
        if (lane == 0) red_sh[w] = mx;
        __syncthreads();
        if (w == 0) {
            float v = red_sh[lane];
            v = fmaxf(v, __shfl_xor(v, 1, 32));
            v = fmaxf(v, __shfl_xor(v, 2, 32));
            v = fmaxf(v, __shfl_xor(v, 4, 32));
            v = fmaxf(v, __shfl_xor(v, 8, 32));
            v = fmaxf(v, __shfl_xor(v, 16, 32));
            if (lane == 0) red_sh[32] = v;
        }
        __syncthreads();
        float p = __expf(x - red_sh[32]);
        float sm = p;
        sm += __shfl_xor(sm, 1, 32);
        sm += __shfl_xor(sm, 2, 32);
        sm += __shfl_xor(sm, 4, 32);
        sm += __shfl_xor(sm, 8, 32);
        sm += __shfl_xor(sm, 16, 32);
        if (lane == 0) red_sh[w] = sm;
        __syncthreads();
        if (w == 0) {
            float v = red_sh[lane];
            v += __shfl_xor(v, 1, 32);
            v += __shfl_xor(v, 2, 32);
            v += __shfl_xor(v, 4, 32);
            v += __shfl_xor(v, 8, 32);
            v += __shfl_xor(v, 16, 32);
            if (lane == 0) red_sh[32] = v;
        }
        __syncthreads();
        p /= red_sh[32];

        // emit e and cumulate alignment
        eo_b[(size_t)step * T_ENC_ + tid] = p;
        align_sh[15 + tid] += p;
        __syncthreads();
    }
}

// ---------------------------------------------------------------------------
// Kernel 2: context GEMM  c[b] = e[b](T_dec x T_enc) @ enc[b](T_enc x E)
// WMMA f32 16x16x4, dual N-tile per wave to reuse A fragments (loaded b64).
// grid = B * (T_dec/16) blocks of 512 threads (16 waves, 32 N-tiles).
// ---------------------------------------------------------------------------
__global__ __launch_bounds__(512)
void lsa_context_kernel(const float* __restrict__ e_out,  // (B, T_dec, T_enc)
                        const float* __restrict__ enc,    // (B, T_enc, E)
                        float* __restrict__ c_out) {      // (B, T_dec, E)
    const int bm   = blockIdx.x;
    const int b    = bm >> 5;             // / (T_DEC_/16)
    const int m0   = (bm & 31) * 16;
    const int w    = threadIdx.x >> 5;    // 0..15
    const int lane = threadIdx.x & 31;
    const int lo   = lane & 15;
    const int hi   = lane >> 4;

    const float* eo_b  = e_out + (size_t)b * T_DEC_ * T_ENC_;
    const float* enc_b = enc + (size_t)b * T_ENC_ * E_;
    float* co_b = c_out + (size_t)b * T_DEC_ * E_;

    const int n0a = 16 * w;
    const int n0b = 16 * w + 256;

    v8f c0 = {};
    v8f c1 = {};
    const float* ap = eo_b + (size_t)(m0 + lo) * T_ENC_;

#pragma unroll 4
    for (int k0 = 0; k0 < T_ENC_; k0 += 4) {
        const int k = k0 + 2 * hi;        // even -> 8B-aligned A fragment
        const v2f a = *reinterpret_cast<const v2f*>(ap + k);
        v2f bb;
        bb.x = enc_b[(size_t)k * E_ + n0a + lo];
        bb.y = enc_b[(size_t)(k + 1) * E_ + n0a + lo];
        c0 = __builtin_amdgcn_wmma_f32_16x16x4_f32(
                 false, a, false, bb, (short)0, c0, false, false);
        bb.x = enc_b[(size_t)k * E_ + n0b + lo];
        bb.y = enc_b[(size_t)(k + 1) * E_ + n0b + lo];
        c1 = __builtin_amdgcn_wmma_f32_16x16x4_f32(
                 false, a, false, bb, (short)0, c1, false, false);
    }

#pragma unroll
    for (int r = 0; r < 8; ++r) {
        const size_t row = (size_t)(m0 + hi * 8 + r) * E_;
        co_b[row + n0a + lo] = c0[r];
        co_b[row + n0b + lo] = c1[r];
    }
}

// ---------------------------------------------------------------------------
extern "C" void kernel_launch(void* const* d_in, const int* in_sizes, int n_in,
                              void* d_out, int out_size, void* d_ws, size_t ws_size,
                              hipStream_t stream) {
    const float* enc    = (const float*)d_in[0];  // (B, T_enc, E)
    const float* dec    = (const float*)d_in[1];  // (B, T_dec, E)
    const float* conv_w = (const float*)d_in[2];  // (31, 1, 32)
    const float* conv_b = (const float*)d_in[3];  // (32)
    const float* w_loc  = (const float*)d_in[4];  // (32, 512)
    const float* v_a    = (const float*)d_in[5];  // (512)
    const float* b_a    = (const float*)d_in[6];  // (512)

    float* out   = (float*)d_out;
    float* c_out = out;                                        // (B,T_dec,E)
    float* e_out = out + (size_t)B_ * T_DEC_ * E_;             // (B,T_dec,T_enc)

    float* wc  = (float*)d_ws;                 // 32*512 floats
    float* cbp = wc + 32 * UNITS_;             // 512 floats

    lsa_prep_kernel<<<1, UNITS_, 0, stream>>>(conv_w, conv_b, w_loc, b_a, wc, cbp);
    lsa_scan_kernel<<<B_, 1024, 0, stream>>>(enc, dec, v_a, wc, cbp, e_out);
    lsa_context_kernel<<<B_ * (T_DEC_ / 16), 512, 0, stream>>>(e_out, enc, c_out);

    (void)in_sizes; (void)n_in; (void)out_size; (void)ws_size;
}